// Peakfinder_26422638805476
// MI455X (gfx1250) — compile-verified
//
#include <hip/hip_runtime.h>
#include <cstdint>

// ---------------------------------------------------------------------------
// Peak finder: 5x5 max-pool (SAME) + peak mask + thresholded scores.
// NHWC (16, 512, 512, 17) f32.  Bandwidth-bound: ~0.86 GB total traffic.
// gfx1250 path: async global->LDS staging (ASYNCcnt + s_wait_asynccnt),
// separable max in LDS, wide non-temporal output stores.
// ---------------------------------------------------------------------------

#define NB       16
#define H        512
#define W        512
#define C        17
#define FW       (W * C)          // 8704 fused (w,c) floats per row
#define TILE_F   512              // fused columns per block (8704 = 17*512)
#define TILE_R   8                // output rows per block
#define ROWS     (TILE_R + 4)     // +2 halo rows each side
#define HALO     34               // 2 pixels * 17 channels
#define LPAD     48               // main-region start inside LDS row (16B aligned)
#define LSTRIDE  608              // floats per LDS row (48 + 512 + 34 + pad)
#define NTHREADS 256

// Builtin signatures (validated by round-1 diagnostics):
//   b32 : (global int* src, local int* dst, imm offset, imm cpol)
//   b128: (global v4i* src, local v4i* dst, imm offset, imm cpol)
typedef int v4i __attribute__((vector_size(16)));
typedef float f4 __attribute__((ext_vector_type(4)));
typedef __attribute__((address_space(1))) int*  g_b32_t;
typedef __attribute__((address_space(3))) int*  l_b32_t;
typedef __attribute__((address_space(1))) v4i*  g_b128_t;
typedef __attribute__((address_space(3))) v4i*  l_b128_t;

#if defined(__has_builtin)
#if __has_builtin(__builtin_amdgcn_global_load_async_to_lds_b32)
#define HAVE_ASYNC_B32 1
#endif
#if __has_builtin(__builtin_amdgcn_global_load_async_to_lds_b128)
#define HAVE_ASYNC_B128 1
#endif
#if __has_builtin(__builtin_amdgcn_s_wait_asynccnt)
#define HAVE_WAIT_ASYNC 1
#endif
#endif

static __device__ __forceinline__ void async_copy_b32(float* lds_dst, const float* gsrc) {
#if defined(HAVE_ASYNC_B32)
    __builtin_amdgcn_global_load_async_to_lds_b32((g_b32_t)gsrc, (l_b32_t)lds_dst, 0, 0);
#else
    *(uint32_t*)lds_dst = *(const uint32_t*)gsrc;
#endif
}

static __device__ __forceinline__ void async_copy_b128(float* lds_dst, const float* gsrc) {
#if defined(HAVE_ASYNC_B128)
    __builtin_amdgcn_global_load_async_to_lds_b128((g_b128_t)gsrc, (l_b128_t)lds_dst, 0, 0);
#else
    *(f4*)lds_dst = *(const f4*)gsrc;
#endif
}

static __device__ __forceinline__ void wait_async_done() {
#if defined(HAVE_WAIT_ASYNC)
    __builtin_amdgcn_s_wait_asynccnt(0);
#elif defined(HAVE_ASYNC_B32) || defined(HAVE_ASYNC_B128)
    asm volatile("s_wait_asynccnt 0" ::: "memory");
#endif
}

__global__ __launch_bounds__(NTHREADS)
void peakfinder_kernel(const float* __restrict__ in,
                       float* __restrict__ out_mask,
                       float* __restrict__ out_score) {
    __shared__ float lin[ROWS][LSTRIDE];   // staged input rows (+ fused halo)
    __shared__ float lrm[ROWS][TILE_F];    // per-row horizontal 5-max

    const int tid = threadIdx.x;
    const int f0  = blockIdx.x * TILE_F;   // fused-column tile origin
    const int h0  = blockIdx.y * TILE_R;   // output row tile origin
    const int n   = blockIdx.z;            // batch

    const float NEG = -__builtin_inff();
    const float* batch = in + (size_t)n * H * FW;

    // ---- Stage 0a: async-load main region (12 rows x 128 float4) ----------
    for (int t = tid; t < ROWS * (TILE_F / 4); t += NTHREADS) {
        const int r = t / (TILE_F / 4);
        const int v = t % (TILE_F / 4);
        const int h = h0 + r - 2;
        float* dst = &lin[r][LPAD + v * 4];
        if ((unsigned)h < (unsigned)H) {
            const float* src = batch + (size_t)h * FW + f0 + v * 4;
            async_copy_b128(dst, src);
        } else {
            *(f4*)dst = (f4){NEG, NEG, NEG, NEG};
        }
    }

    // ---- Stage 0b: fused-dim halos (34 floats each side per row) ----------
    for (int t = tid; t < ROWS * HALO; t += NTHREADS) {
        const int r = t / HALO;
        const int j = t % HALO;
        const int h = h0 + r - 2;
        // left halo
        {
            float* dst = &lin[r][LPAD - HALO + j];
            const int gf = f0 - HALO + j;
            if ((unsigned)h < (unsigned)H && gf >= 0)
                async_copy_b32(dst, batch + (size_t)h * FW + gf);
            else
                *dst = NEG;
        }
        // right halo
        {
            float* dst = &lin[r][LPAD + TILE_F + j];
            const int gf = f0 + TILE_F + j;
            if ((unsigned)h < (unsigned)H && gf < FW)
                async_copy_b32(dst, batch + (size_t)h * FW + gf);
            else
                *dst = NEG;
        }
    }

    wait_async_done();
    __syncthreads();

    // ---- Stage 1: horizontal 5-max (offsets 0, +-17, +-34 in fused dim) ---
    // Scalar reads (unaligned +-17 offsets), vectorized b128 write per 4 outputs.
    for (int t = tid; t < ROWS * (TILE_F / 4); t += NTHREADS) {
        const int r = t / (TILE_F / 4);
        const int i = (t % (TILE_F / 4)) * 4;
        f4 m;
#pragma unroll
        for (int k = 0; k < 4; ++k) {
            const float* p = &lin[r][LPAD + i + k];
            float v = p[0];
            v = fmaxf(v, p[-C]);
            v = fmaxf(v, p[+C]);
            v = fmaxf(v, p[-2 * C]);
            v = fmaxf(v, p[+2 * C]);
            m[k] = v;
        }
        *(f4*)&lrm[r][i] = m;
    }
    __syncthreads();

    // ---- Stage 2: vertical 5-max, peak test, b128 non-temporal stores -----
    for (int t = tid; t < TILE_R * (TILE_F / 4); t += NTHREADS) {
        const int rr = t / (TILE_F / 4);
        const int i  = (t % (TILE_F / 4)) * 4;
        f4 m = *(const f4*)&lrm[rr][i];
#pragma unroll
        for (int k = 1; k < 5; ++k)
            m = __builtin_elementwise_max(m, *(const f4*)&lrm[rr + k][i]);
        const f4 center = *(const f4*)&lin[rr + 2][LPAD + i];
        f4 maskv, scorev;
#pragma unroll
        for (int k = 0; k < 4; ++k) {
            const bool valid = (m[k] == center[k]) && (center[k] > 0.01f);
            maskv[k]  = valid ? 1.0f : 0.0f;
            scorev[k] = valid ? center[k] : 0.0f;
        }
        const size_t flat = ((size_t)(n * H + h0 + rr)) * FW + f0 + i;
        __builtin_nontemporal_store(maskv,  (f4*)&out_mask[flat]);
        __builtin_nontemporal_store(scorev, (f4*)&out_score[flat]);
    }
}

extern "C" void kernel_launch(void* const* d_in, const int* in_sizes, int n_in,
                              void* d_out, int out_size, void* d_ws, size_t ws_size,
                              hipStream_t stream) {
    (void)in_sizes; (void)n_in; (void)out_size; (void)d_ws; (void)ws_size;
    const float* in = (const float*)d_in[0];
    float* out = (float*)d_out;
    const size_t TOTAL = (size_t)NB * H * FW;   // 71,303,168 elements per output
    dim3 grid(FW / TILE_F, H / TILE_R, NB);     // (17, 64, 16)
    peakfinder_kernel<<<grid, NTHREADS, 0, stream>>>(in, out, out + TOTAL);
}